// RaycastRGBD_39934605919044
// MI455X (gfx1250) — compile-verified
//
#include <hip/hip_runtime.h>
#include <hip/hip_bf16.h>
#include <stdint.h>

#define Bc 2
#define DZc 64
#define DYc 96
#define DXc 96
#define Hc 192
#define Wc 256
#define NSTEPSc 96
#define DEPTH_MINc 1.0f
#define RAY_INCc 1.0f
#define THRESHc 3.0f
#define TRUNCc 3.0f

static constexpr int    kHW   = Hc * Wc;                      // 49152
static constexpr int    kBHW  = Bc * kHW;                     // 98304
static constexpr size_t kNvox = (size_t)Bc * DZc * DYc * DXc; // 1179648

// 2-float vector with alignment relaxed to 4 so (x0, x0+1) corner pairs can be
// fetched with one global_load_b64 regardless of x0 parity.
typedef float vfloat2 __attribute__((ext_vector_type(2)));
typedef vfloat2 __attribute__((aligned(4))) vfloat2_a4;

// WMMA operand types (CDNA5 V_WMMA_F32_16X16X4_F32: A,B = 2 VGPR f32; C,D = 8 VGPR f32)
typedef float v2f __attribute__((ext_vector_type(2)));
typedef float v8f __attribute__((ext_vector_type(8)));

// ---------------------------------------------------------------- init grid
__global__ __launch_bounds__(256) void init_grid_k(float* __restrict__ grid,
                                                   float* __restrict__ sdf,
                                                   int compact) {
    size_t i = (size_t)blockIdx.x * blockDim.x + threadIdx.x;
    if (i >= kNvox) return;
    float4* g = (float4*)(grid + i * 8);
    g[0] = make_float4(TRUNCc, 0.f, 0.f, 0.f);
    g[1] = make_float4(0.f, 0.f, 0.f, 0.f);
    if (compact) sdf[i] = TRUNCc;
}

// ---------------------------------------------------------------- scatter
__global__ __launch_bounds__(256) void scatter_k(const int* __restrict__ locs,
                                                 const float* __restrict__ vs,
                                                 const float* __restrict__ vc,
                                                 const float* __restrict__ vn,
                                                 float* __restrict__ grid,
                                                 float* __restrict__ sdf,
                                                 int compact, int n) {
    int i = blockIdx.x * blockDim.x + threadIdx.x;
    if (i >= n) return;
    int z = __builtin_nontemporal_load(&locs[4 * i + 0]);
    int y = __builtin_nontemporal_load(&locs[4 * i + 1]);
    int x = __builtin_nontemporal_load(&locs[4 * i + 2]);
    int b = __builtin_nontemporal_load(&locs[4 * i + 3]);
    size_t v = (((size_t)b * DZc + z) * DYc + y) * DXc + x;
    float s  = __builtin_nontemporal_load(&vs[i]);
    float c0 = __builtin_nontemporal_load(&vc[3 * i + 0]);
    float c1 = __builtin_nontemporal_load(&vc[3 * i + 1]);
    float c2 = __builtin_nontemporal_load(&vc[3 * i + 2]);
    float n0 = __builtin_nontemporal_load(&vn[3 * i + 0]);
    float n1 = __builtin_nontemporal_load(&vn[3 * i + 1]);
    float n2 = __builtin_nontemporal_load(&vn[3 * i + 2]);
    float4* g = (float4*)(grid + v * 8);
    g[0] = make_float4(s, c0, c1, c2);
    g[1] = make_float4(n0, n1, n2, 1.0f);
    if (compact) sdf[v] = s;
}

// ---------------------------------------------------------------- trilinear SDF
template <int SS>
__device__ __forceinline__ float tri_sdf_t(const float* __restrict__ sdf, size_t bbase,
                                           float px, float py, float pz, size_t* basep) {
    float x = fminf(fmaxf(px, 0.f), (float)(DXc - 1) - 1e-4f);
    float y = fminf(fmaxf(py, 0.f), (float)(DYc - 1) - 1e-4f);
    float z = fminf(fmaxf(pz, 0.f), (float)(DZc - 1) - 1e-4f);
    int x0 = (int)x, y0 = (int)y, z0 = (int)z;   // clamped >= 0: trunc == floor
    float wx = x - (float)x0, wy = y - (float)y0, wz = z - (float)z0;
    size_t base = bbase + ((size_t)z0 * DYc + y0) * DXc + x0;
    *basep = base;
    const float* p = sdf + base * (size_t)SS;
    const int sy = DXc * SS, sz = DYc * DXc * SS;
    float c00, c01, c10, c11;
    if (SS == 1) {
        // adjacent x-pairs: 4x global_load_b64 instead of 8x b32
        vfloat2 a = *(const vfloat2_a4*)(p);
        vfloat2 b = *(const vfloat2_a4*)(p + sy);
        vfloat2 c = *(const vfloat2_a4*)(p + sz);
        vfloat2 d = *(const vfloat2_a4*)(p + sz + sy);
        c00 = a.x + wx * (a.y - a.x);
        c01 = b.x + wx * (b.y - b.x);
        c10 = c.x + wx * (c.y - c.x);
        c11 = d.x + wx * (d.y - d.x);
    } else {
        float c000 = p[0],        c001 = p[SS];
        float c010 = p[sy],       c011 = p[sy + SS];
        float c100 = p[sz],       c101 = p[sz + SS];
        float c110 = p[sz + sy],  c111 = p[sz + sy + SS];
        c00 = c000 + wx * (c001 - c000);
        c01 = c010 + wx * (c011 - c010);
        c10 = c100 + wx * (c101 - c100);
        c11 = c110 + wx * (c111 - c110);
    }
    float c0 = c00 + wy * (c01 - c00);
    float c1 = c10 + wy * (c11 - c10);
    return c0 + wz * (c1 - c0);
}

// ---------------------------------------------------------------- trilinear 8-channel
__device__ __forceinline__ void tri_attr(const float* __restrict__ grid, size_t bbase,
                                         float px, float py, float pz, float out[8]) {
    float x = fminf(fmaxf(px, 0.f), (float)(DXc - 1) - 1e-4f);
    float y = fminf(fmaxf(py, 0.f), (float)(DYc - 1) - 1e-4f);
    float z = fminf(fmaxf(pz, 0.f), (float)(DZc - 1) - 1e-4f);
    int x0 = (int)x, y0 = (int)y, z0 = (int)z;
    float wx = x - (float)x0, wy = y - (float)y0, wz = z - (float)z0;
    size_t base = bbase + ((size_t)z0 * DYc + y0) * DXc + x0;
    const float* p = grid + base * 8;
    const size_t SX = 8, SY = (size_t)DXc * 8, SZ = (size_t)DYc * DXc * 8;
#pragma unroll
    for (int k = 0; k < 8; ++k) out[k] = 0.f;
#pragma unroll
    for (int dz = 0; dz < 2; ++dz)
#pragma unroll
        for (int dy = 0; dy < 2; ++dy)
#pragma unroll
            for (int dx = 0; dx < 2; ++dx) {
                float wgt = (dz ? wz : 1.f - wz) * (dy ? wy : 1.f - wy) * (dx ? wx : 1.f - wx);
                const float* q = p + (size_t)dz * SZ + (size_t)dy * SY + (size_t)dx * SX;
                float4 q0 = *(const float4*)q;
                float4 q1 = *(const float4*)(q + 4);
                out[0] += wgt * q0.x; out[1] += wgt * q0.y;
                out[2] += wgt * q0.z; out[3] += wgt * q0.w;
                out[4] += wgt * q1.x; out[5] += wgt * q1.y;
                out[6] += wgt * q1.z; out[7] += wgt * q1.w;
            }
}

// ---------------------------------------------------------------- raycast
template <int SS>
__global__ __launch_bounds__(256) void raycast_k(const float* __restrict__ grid,
                                                 const float* __restrict__ sdf,
                                                 const float* __restrict__ view,
                                                 const float* __restrict__ intr,
                                                 float* __restrict__ out_color,
                                                 float* __restrict__ out_depth,
                                                 float* __restrict__ out_normal) {
    int p = blockIdx.x * blockDim.x + threadIdx.x;
    if (p >= kBHW) return;                 // never taken: grid exactly covers kBHW
    int b = p / kHW;                       // uniform per 256-thread block
    int r = p % kHW;
    int h = r / Wc;
    int w = r % Wc;

    __shared__ float vm[16];
#if defined(__gfx1250__)
    // CDNA5 async global->LDS broadcast of the per-batch 4x4 view matrix.
    if (threadIdx.x < 16) {
        unsigned long long ga = (unsigned long long)(const void*)(view + b * 16 + threadIdx.x);
        unsigned int       lo = (unsigned int)(uintptr_t)(&vm[threadIdx.x]); // low 32 bits = LDS offset
        asm volatile("global_load_async_to_lds_b32 %0, %1, off" :: "v"(lo), "v"(ga) : "memory");
    }
    asm volatile("s_wait_asynccnt 0" ::: "memory");
    __syncthreads();
#else
    if (threadIdx.x < 16) vm[threadIdx.x] = view[b * 16 + threadIdx.x];
    __syncthreads();
#endif

    float fx = intr[0], fy = intr[1], mx = intr[2], my = intr[3];
    float dcx = ((float)w - mx) / fx;
    float dcy = ((float)h - my) / fy;

    // world ray: pos(t) = origin + t * dw,  dw = R * dir_cam
    float dwx, dwy, dwz;
#if defined(__gfx1250__) && __has_builtin(__builtin_amdgcn_wmma_f32_16x16x4_f32)
    {
        // D(16x16) = A(16x4) x B(4x16): A rows 0..2 = R (K=3 col zero), B col n =
        // (dcx_n, dcy_n, 1, 0). D rows 0..2 = (dwx, dwy, dwz) per pixel. Two WMMAs
        // cover the wave's 32 pixels (16 columns each).
        int  lane = threadIdx.x & 31;      // wave32
        int  half = lane & 15;
        bool hi   = lane >= 16;
        // A layout (16x4 f32): lanes 0-15 hold {A[m][0],A[m][1]} (m=lane),
        // lanes 16-31 hold {A[m][2],A[m][3]} (m=lane-16).
        int   mi = (half < 3) ? half : 0;
        float ax = vm[4 * mi + (hi ? 2 : 0)];
        float ay = hi ? 0.f : vm[4 * mi + 1];
        if (half >= 3) { ax = 0.f; ay = 0.f; }
        v2f A = {ax, ay};
        // B layout (4x16 f32): lanes 0-15 hold {B[0][n],B[1][n]} (n=lane),
        // lanes 16-31 hold {B[2][n],B[3][n]} (n=lane-16).
        v2f B1 = {hi ? 1.f : dcx, hi ? 0.f : dcy};                    // pixels 0..15
        float dcx_hi = __shfl(dcx, half + 16, 32);
        float dcy_hi = __shfl(dcy, half + 16, 32);
        v2f B2 = {hi ? 1.f : dcx_hi, hi ? 0.f : dcy_hi};              // pixels 16..31
        v8f cz = {0.f, 0.f, 0.f, 0.f, 0.f, 0.f, 0.f, 0.f};
        v8f D1 = __builtin_amdgcn_wmma_f32_16x16x4_f32(false, A, false, B1,
                                                       (short)0, cz, false, false);
        v8f D2 = __builtin_amdgcn_wmma_f32_16x16x4_f32(false, A, false, B2,
                                                       (short)0, cz, false, false);
        // D layout: VGPR m, lanes 0-15 = row M=m, col N=lane. Rows 0..2 = dw.
        float w0 = __shfl(D2[0], half, 32);
        float w1 = __shfl(D2[1], half, 32);
        float w2 = __shfl(D2[2], half, 32);
        dwx = hi ? w0 : D1[0];
        dwy = hi ? w1 : D1[1];
        dwz = hi ? w2 : D1[2];
    }
#else
    dwx = vm[0] * dcx + vm[1] * dcy + vm[2];
    dwy = vm[4] * dcx + vm[5] * dcy + vm[6];
    dwz = vm[8] * dcx + vm[9] * dcy + vm[10];
#endif
    float ox = vm[3], oy = vm[7], oz = vm[11];

    size_t bbase = (size_t)b * DZc * DYc * DXc;

    // Per-ray approximate voxel advance per step (direction is constant), used
    // for a one-add-per-step prefetch of the NEXT step's base cell.
    int sox = (int)floorf(dwx + 0.5f);
    int soy = (int)floorf(dwy + 0.5f);
    int soz = (int)floorf(dwz + 0.5f);
    long long soff2 = 2LL * ((long long)soz * (DYc * DXc) + (long long)soy * DXc + sox);

    float px = ox + DEPTH_MINc * dwx;
    float py = oy + DEPTH_MINc * dwy;
    float pz = oz + DEPTH_MINc * dwz;
    size_t base;
    float prev = tri_sdf_t<SS>(sdf, bbase, px, py, pz, &base);

    int hit = -1;
    float sp = 0.f, sc = 0.f;
    for (int i = 0; i < NSTEPSc - 1; ++i) {
        px += dwx; py += dwy; pz += dwz;
        // prefetch ~next step's base corner (global_prefetch_b8), clamped in-range
        long long pf = (long long)base + soff2;
        pf = pf < 0 ? 0 : (pf > (long long)(kNvox - 1) ? (long long)(kNvox - 1) : pf);
        __builtin_prefetch(sdf + (size_t)pf * (size_t)SS, 0, 1);
        float cur = tri_sdf_t<SS>(sdf, bbase, px, py, pz, &base);
        if (prev > 0.f && cur <= 0.f && fabsf(prev) < THRESHc) {
            hit = i; sp = prev; sc = cur;
            break;
        }
        prev = cur;
    }

    float oc0 = 0.f, oc1 = 0.f, oc2 = 0.f, od = 0.f, on0 = 0.f, on1 = 0.f, on2 = 0.f;
    if (hit >= 0) {
        float denom = fmaxf(sp - sc, 1e-8f);
        float frac  = fminf(fmaxf(sp / denom, 0.f), 1.f);
        float t     = DEPTH_MINc + ((float)hit + frac) * RAY_INCc;
        float hx = ox + t * dwx, hy = oy + t * dwy, hz = oz + t * dwz;
        float a[8];
        tri_attr(grid, bbase, hx, hy, hz, a);
        float inv = 1.f / fmaxf(a[7], 1e-5f);
        oc0 = a[1] * inv; oc1 = a[2] * inv; oc2 = a[3] * inv;
        float n0 = a[4] * inv, n1 = a[5] * inv, n2 = a[6] * inv;
        float nn  = sqrtf(n0 * n0 + n1 * n1 + n2 * n2);
        float in2 = 1.f / fmaxf(nn, 1e-8f);
        on0 = n0 * in2; on1 = n1 * in2; on2 = n2 * in2;
        od = t;
    }
    // streaming (non-temporal) stores: outputs are never re-read on device
    __builtin_nontemporal_store(oc0, &out_color[3 * p + 0]);
    __builtin_nontemporal_store(oc1, &out_color[3 * p + 1]);
    __builtin_nontemporal_store(oc2, &out_color[3 * p + 2]);
    __builtin_nontemporal_store(od,  &out_depth[p]);
    __builtin_nontemporal_store(on0, &out_normal[3 * p + 0]);
    __builtin_nontemporal_store(on1, &out_normal[3 * p + 1]);
    __builtin_nontemporal_store(on2, &out_normal[3 * p + 2]);
}

// ---------------------------------------------------------------- launch
extern "C" void kernel_launch(void* const* d_in, const int* in_sizes, int n_in,
                              void* d_out, int out_size, void* d_ws, size_t ws_size,
                              hipStream_t stream) {
    const int*   locs = (const int*)d_in[0];
    const float* vs   = (const float*)d_in[1];
    const float* vc   = (const float*)d_in[2];
    const float* vn   = (const float*)d_in[3];
    const float* view = (const float*)d_in[4];
    const float* intr = (const float*)d_in[5];
    int n = in_sizes[0] / 4;

    float* grid = (float*)d_ws;
    size_t grid_bytes = kNvox * 8 * sizeof(float);
    size_t sdf_bytes  = kNvox * sizeof(float);
    int    compact = (ws_size >= grid_bytes + sdf_bytes) ? 1 : 0;
    float* sdfp = compact ? (grid + kNvox * 8) : grid;

    init_grid_k<<<(int)((kNvox + 255) / 256), 256, 0, stream>>>(grid, sdfp, compact);
    scatter_k<<<(n + 255) / 256, 256, 0, stream>>>(locs, vs, vc, vn, grid, sdfp, compact, n);

    float* out  = (float*)d_out;
    float* ocol = out;                          // B*H*W*3
    float* odep = out + (size_t)kBHW * 3;       // B*H*W
    float* onrm = odep + (size_t)kBHW;          // B*H*W*3
    if (compact) {
        raycast_k<1><<<kBHW / 256, 256, 0, stream>>>(grid, sdfp, view, intr, ocol, odep, onrm);
    } else {
        raycast_k<8><<<kBHW / 256, 256, 0, stream>>>(grid, sdfp, view, intr, ocol, odep, onrm);
    }
}